// DecoderLSTM_22823456211767
// MI455X (gfx1250) — compile-verified
//
#include <hip/hip_runtime.h>
#include <hip/hip_bf16.h>
#include <math.h>
#include <stdint.h>

// Problem constants (match reference)
#define Bsz 128
#define Tt  24
#define Vv  10000
#define Ee  512
#define Hh  512
#define Ff  2048
#define Aa  512
#define Pp  196

typedef float v2f __attribute__((ext_vector_type(2)));
typedef float v8f __attribute__((ext_vector_type(8)));

// Async global->LDS copy of 16B per lane (CDNA5 GLOBAL_LOAD_ASYNC_TO_LDS_B128,
// tracked by ASYNCcnt). lp = LDS byte address (low 32 bits of a generic pointer
// to LDS are the LDS offset per the aperture rules), gp = per-lane global addr.
__device__ __forceinline__ void async_copy_b128(uint32_t lp, const float* gp) {
    asm volatile("global_load_async_to_lds_b128 %0, %1, off"
                 :: "v"(lp), "v"((unsigned long long)(uintptr_t)gp)
                 : "memory");
}
__device__ __forceinline__ void wait_async() {
    asm volatile("s_wait_asynccnt 0x0" ::: "memory");
}
__device__ __forceinline__ uint32_t lds_addr(const void* p) {
    return (uint32_t)(uintptr_t)p;
}

// ---------------------------------------------------------------------------
// Generic fp32 WMMA GEMM:  C[M,N] = act(A[M,K] * op(B) + bias)
//  BNT=false: B stored row-major K x N (ldb = row stride)
//  BNT=true : B stored row-major N x K (ldb = row stride)  => C = A * B^T
// Block tile 64x64, 4 waves, each wave -> 16 rows x 64 cols via
// V_WMMA_F32_16X16X4_F32 (ISA 7.12 f32 operand layouts).
// LDS strides: A rows = 20 floats (80B, 16B aligned); B-NN rows = 68 floats;
// B-NT rows = 20 floats -> contiguous (k,k+1) pairs for single b64 frag loads.
// Interior tiles: DOUBLE-BUFFERED async b128 staging -- tile i+1 streams into
// LDS (ASYNCcnt) while tile i's 16 WMMAs execute; one barrier per K-tile.
// Ragged tiles (fc N=10000 edge) use a guarded scalar path (blockIdx-uniform).
// ---------------------------------------------------------------------------
template<bool BNT, bool RELU>
__global__ __launch_bounds__(128) void gemm_wmma_f32(
    const float* __restrict__ A, int lda,
    const float* __restrict__ Bm, int ldb,
    const float* __restrict__ bias,
    float* __restrict__ C, int ldc,
    int M, int N, int K)
{
    __shared__ float As[2][64 * 20];
    __shared__ float Bs[2][1280];       // per buffer: max(16*68, 64*20)

    const int tid  = threadIdx.x;
    const int lane = tid & 31;
    const int wave = tid >> 5;
    const int m0   = blockIdx.y * 64;
    const int n0   = blockIdx.x * 64;
    const bool full = (m0 + 64 <= M) && (n0 + 64 <= N);

    v8f acc[4];
    for (int j = 0; j < 4; ++j)
        for (int r = 0; r < 8; ++r) acc[j][r] = 0.0f;

    const int mrow = wave * 16;
    const int ml   = lane & 15;
    const int koff = (lane >> 4) * 2;   // lanes 0-15: K 0/1, lanes 16-31: K 2/3

    // 16 WMMAs on the K=16 tile sitting in LDS buffer bi
    auto compute = [&](int bi) {
        const float* as = As[bi];
        const float* bs = Bs[bi];
        #pragma unroll
        for (int k4 = 0; k4 < 16; k4 += 4) {
            v2f a;
            a.x = as[(mrow + ml) * 20 + k4 + koff];
            a.y = as[(mrow + ml) * 20 + k4 + koff + 1];
            #pragma unroll
            for (int j = 0; j < 4; ++j) {
                v2f bfr;
                if (BNT) {
                    bfr.x = bs[(j * 16 + ml) * 20 + k4 + koff];
                    bfr.y = bs[(j * 16 + ml) * 20 + k4 + koff + 1];
                } else {
                    bfr.x = bs[(k4 + koff) * 68 + j * 16 + ml];
                    bfr.y = bs[(k4 + koff + 1) * 68 + j * 16 + ml];
                }
                acc[j] = __builtin_amdgcn_wmma_f32_16x16x4_f32(
                    false, a, false, bfr, (short)0, acc[j], false, false);
            }
        }
    };

    if (full) {
        // issue async DMA of one K-tile into buffer bi
        auto stage = [&](int bi, int k0) {
            #pragma unroll
            for (int i = 0; i < 2; ++i) {                   // A: 64x16
                int idx = tid + i * 128;
                int r = idx >> 2;                           // 0..63
                int q = idx & 3;                            // x4 floats
                async_copy_b128(lds_addr(&As[bi][r * 20 + q * 4]),
                                A + (size_t)(m0 + r) * lda + k0 + q * 4);
            }
            if (BNT) {                                      // LDS [n][k]
                #pragma unroll
                for (int i = 0; i < 2; ++i) {
                    int idx = tid + i * 128;
                    int n = idx >> 2;
                    int q = idx & 3;
                    async_copy_b128(lds_addr(&Bs[bi][n * 20 + q * 4]),
                                    Bm + (size_t)(n0 + n) * ldb + k0 + q * 4);
                }
            } else {                                        // LDS [k][n]
                #pragma unroll
                for (int i = 0; i < 2; ++i) {
                    int idx = tid + i * 128;
                    int r = idx >> 4;                       // 0..15
                    int q = idx & 15;                       // x4 floats
                    async_copy_b128(lds_addr(&Bs[bi][r * 68 + q * 4]),
                                    Bm + (size_t)(k0 + r) * ldb + n0 + q * 4);
                }
            }
        };

        stage(0, 0);
        int cur = 0;
        for (int k0 = 0; k0 < K; k0 += 16) {
            wait_async();           // tile `cur` resident (this wave's DMAs)
            __syncthreads();        // all waves' DMAs done; all waves done reading cur^1
            if (k0 + 16 < K) stage(cur ^ 1, k0 + 16);   // overlap next DMA with compute
            compute(cur);
            cur ^= 1;
        }
    } else {
        // ---- guarded scalar staging (ragged tiles), single buffer ----
        for (int k0 = 0; k0 < K; k0 += 16) {
            #pragma unroll
            for (int i = 0; i < 8; ++i) {
                int idx = tid + i * 128;   // 0..1023
                int r = idx >> 4;          // 0..63
                int c = idx & 15;
                int gm = m0 + r;
                As[0][r * 20 + c] = (gm < M) ? A[(size_t)gm * lda + (k0 + c)] : 0.0f;
            }
            if (BNT) {
                #pragma unroll
                for (int i = 0; i < 8; ++i) {
                    int idx = tid + i * 128;
                    int n = idx >> 4;      // 0..63
                    int k = idx & 15;
                    int gn = n0 + n;
                    Bs[0][n * 20 + k] = (gn < N) ? Bm[(size_t)gn * ldb + (k0 + k)] : 0.0f;
                }
            } else {
                #pragma unroll
                for (int i = 0; i < 8; ++i) {
                    int idx = tid + i * 128;
                    int r = idx >> 6;      // 0..15
                    int c = idx & 63;
                    int gn = n0 + c;
                    Bs[0][r * 68 + c] = (gn < N) ? Bm[(size_t)(k0 + r) * ldb + gn] : 0.0f;
                }
            }
            __syncthreads();
            compute(0);
            __syncthreads();
        }
    }

    // --- write out: C/D layout VGPR r -> row r (lanes 0-15) / r+8 (lanes 16-31)
    const int mbase = m0 + wave * 16 + ((lane >> 4) << 3);
    #pragma unroll
    for (int j = 0; j < 4; ++j) {
        int gn = n0 + j * 16 + ml;
        if (gn >= N) continue;
        float bv = bias ? bias[gn] : 0.0f;
        #pragma unroll
        for (int r = 0; r < 8; ++r) {
            int gm = mbase + r;
            if (gm < M) {
                float v = acc[j][r] + bv;
                if (RELU) v = fmaxf(v, 0.0f);
                C[(size_t)gm * ldc + gn] = v;
            }
        }
    }
}

// ---------------------------------------------------------------------------
// Attention scores: one wave per (b,p): s = att_v . tanh(feat_att[b,p,:] + hWh[b,:]) + bv
// ---------------------------------------------------------------------------
__global__ __launch_bounds__(128) void att_scores_kernel(
    const float* __restrict__ feat_att, const float* __restrict__ hWh,
    const float* __restrict__ att_v, const float* __restrict__ att_bv,
    float* __restrict__ scores)
{
    int wid  = (blockIdx.x * 128 + threadIdx.x) >> 5;
    int lane = threadIdx.x & 31;
    if (wid >= Bsz * Pp) return;
    int b = wid / Pp;
    const float* fa = feat_att + (size_t)wid * Aa;
    const float* hw = hWh + (size_t)b * Aa;
    float s = 0.0f;
    for (int j = lane; j < Aa; j += 32)
        s += att_v[j] * tanhf(fa[j] + hw[j]);
    for (int off = 16; off > 0; off >>= 1)
        s += __shfl_down(s, off, 32);
    if (lane == 0) scores[wid] = s + att_bv[0];
}

// ---------------------------------------------------------------------------
// Softmax over P + weighted context: ctx[b,f] = sum_p alpha[b,p] * features[b,p,f]
// ---------------------------------------------------------------------------
__global__ __launch_bounds__(256) void softmax_ctx_kernel(
    const float* __restrict__ scores, const float* __restrict__ features,
    float* __restrict__ ctx)
{
    __shared__ float red[256];
    __shared__ float alpha[Pp];
    int b = blockIdx.x, tid = threadIdx.x;

    float sv = (tid < Pp) ? scores[b * Pp + tid] : -3.4e38f;
    red[tid] = sv; __syncthreads();
    for (int o = 128; o > 0; o >>= 1) {
        if (tid < o) red[tid] = fmaxf(red[tid], red[tid + o]);
        __syncthreads();
    }
    float mx = red[0]; __syncthreads();
    float e = (tid < Pp) ? __expf(sv - mx) : 0.0f;
    red[tid] = e; __syncthreads();
    for (int o = 128; o > 0; o >>= 1) {
        if (tid < o) red[tid] += red[tid + o];
        __syncthreads();
    }
    float inv = 1.0f / red[0];
    if (tid < Pp) alpha[tid] = e * inv;
    __syncthreads();

    const float* fb = features + (size_t)b * Pp * Ff;
    for (int f = tid; f < Ff; f += 256) {
        float acc = 0.0f;
        for (int p = 0; p < Pp; ++p)
            acc = fmaf(alpha[p], fb[(size_t)p * Ff + f], acc);
        ctx[(size_t)b * Ff + f] = acc;
    }
}

// ---------------------------------------------------------------------------
// LSTM pointwise update (PyTorch gate order i,f,g,o)
// ---------------------------------------------------------------------------
__global__ void lstm_update_kernel(const float* __restrict__ gates,
                                   float* __restrict__ h, float* __restrict__ c)
{
    int idx = blockIdx.x * blockDim.x + threadIdx.x;   // b*H + j
    if (idx >= Bsz * Hh) return;
    int b = idx / Hh, j = idx % Hh;
    const float* g = gates + (size_t)b * 4 * Hh;
    float gi = g[j], gf = g[Hh + j], gg = g[2 * Hh + j], go = g[3 * Hh + j];
    float si = 1.0f / (1.0f + __expf(-gi));
    float sf = 1.0f / (1.0f + __expf(-gf));
    float so = 1.0f / (1.0f + __expf(-go));
    float c2 = sf * c[idx] + si * tanhf(gg);
    c[idx] = c2;
    h[idx] = so * tanhf(c2);
}

// ---------------------------------------------------------------------------
// LayerNorm over H=512, block per batch row, 256 threads x 2 elems
// ---------------------------------------------------------------------------
__global__ __launch_bounds__(256) void layernorm_kernel(
    const float* __restrict__ x, const float* __restrict__ g,
    const float* __restrict__ bta, float* __restrict__ y)
{
    __shared__ float red[256];
    int b = blockIdx.x, tid = threadIdx.x;
    const float* xb = x + (size_t)b * Hh;
    float v0 = xb[tid], v1 = xb[tid + 256];
    red[tid] = v0 + v1; __syncthreads();
    for (int o = 128; o > 0; o >>= 1) {
        if (tid < o) red[tid] += red[tid + o];
        __syncthreads();
    }
    float mu = red[0] * (1.0f / Hh); __syncthreads();
    float d0 = v0 - mu, d1 = v1 - mu;
    red[tid] = d0 * d0 + d1 * d1; __syncthreads();
    for (int o = 128; o > 0; o >>= 1) {
        if (tid < o) red[tid] += red[tid + o];
        __syncthreads();
    }
    float inv = rsqrtf(red[0] * (1.0f / Hh) + 1e-5f);
    y[(size_t)b * Hh + tid]       = d0 * inv * g[tid]       + bta[tid];
    y[(size_t)b * Hh + tid + 256] = d1 * inv * g[tid + 256] + bta[tid + 256];
}

// ---------------------------------------------------------------------------
// Concats (embedding gathered in place; avoids a separate embs buffer)
// ---------------------------------------------------------------------------
__global__ void build_xh0_kernel(const int* __restrict__ captions, int t,
    const float* __restrict__ emb, const float* __restrict__ ctxp,
    const float* __restrict__ h0, float* __restrict__ xh0)
{
    int idx = blockIdx.x * blockDim.x + threadIdx.x;
    if (idx >= Bsz * 1536) return;
    int b = idx / 1536, col = idx % 1536;
    float v;
    if      (col < Ee)      v = emb[(size_t)captions[b * Tt + t] * Ee + col];
    else if (col < Ee + Hh) v = ctxp[b * Hh + (col - Ee)];
    else                    v = h0[b * Hh + (col - Ee - Hh)];
    xh0[idx] = v;
}

__global__ void build_xh1_kernel(const float* __restrict__ h0,
    const float* __restrict__ h1, float* __restrict__ xh1)
{
    int idx = blockIdx.x * blockDim.x + threadIdx.x;
    if (idx >= Bsz * 1024) return;
    int b = idx / 1024, col = idx % 1024;
    xh1[idx] = (col < Hh) ? h0[b * Hh + col] : h1[b * Hh + (col - Hh)];
}

__global__ void build_skipin_kernel(const float* __restrict__ h1,
    const int* __restrict__ captions, int t, const float* __restrict__ emb,
    const float* __restrict__ ctxp, float* __restrict__ skipin)
{
    int idx = blockIdx.x * blockDim.x + threadIdx.x;
    if (idx >= Bsz * 1536) return;
    int b = idx / 1536, col = idx % 1536;
    float v;
    if      (col < Hh)      v = h1[b * Hh + col];
    else if (col < Hh + Ee) v = emb[(size_t)captions[b * Tt + t] * Ee + (col - Hh)];
    else                    v = ctxp[b * Hh + (col - Hh - Ee)];
    skipin[idx] = v;
}

// Pack [Wa | Wb] along K so x@W_ih.T + h@W_hh.T becomes one NT GEMM.
__global__ void pack_cat_kernel(const float* __restrict__ Wa, int Ka,
    const float* __restrict__ Wb, int Kb, float* __restrict__ out, int Ntot)
{
    int Kc = Ka + Kb;
    int idx = blockIdx.x * blockDim.x + threadIdx.x;
    if (idx >= Ntot * Kc) return;
    int n = idx / Kc, k = idx % Kc;
    out[idx] = (k < Ka) ? Wa[(size_t)n * Ka + k] : Wb[(size_t)n * Kb + (k - Ka)];
}

__global__ void add_vec_kernel(const float* __restrict__ a,
    const float* __restrict__ b, float* __restrict__ o, int n)
{
    int i = blockIdx.x * blockDim.x + threadIdx.x;
    if (i < n) o[i] = a[i] + b[i];
}

// ---------------------------------------------------------------------------
extern "C" void kernel_launch(void* const* d_in, const int* in_sizes, int n_in,
                              void* d_out, int out_size, void* d_ws, size_t ws_size,
                              hipStream_t stream) {
    const float* features = (const float*)d_in[0];
    const int*   captions = (const int*)  d_in[1];
    const float* emb      = (const float*)d_in[2];
    const float* att_Wf   = (const float*)d_in[3];
    const float* att_bf   = (const float*)d_in[4];
    const float* att_Wh   = (const float*)d_in[5];
    const float* att_bh   = (const float*)d_in[6];
    const float* att_v    = (const float*)d_in[7];
    const float* att_bv   = (const float*)d_in[8];
    const float* proj_W   = (const float*)d_in[9];
    const float* proj_b   = (const float*)d_in[10];
    const float* W_ih0    = (const float*)d_in[11];
    const float* W_hh0    = (const float*)d_in[12];
    const float* b_ih0    = (const float*)d_in[13];
    const float* b_hh0    = (const float*)d_in[14];
    const float* W_ih1    = (const float*)d_in[15];
    const float* W_hh1    = (const float*)d_in[16];
    const float* b_ih1    = (const float*)d_in[17];
    const float* b_hh1    = (const float*)d_in[18];
    const float* ln_g     = (const float*)d_in[19];
    const float* ln_b     = (const float*)d_in[20];
    const float* skip_W   = (const float*)d_in[21];
    const float* skip_b   = (const float*)d_in[22];
    const float* fc_W     = (const float*)d_in[23];
    const float* fc_b     = (const float*)d_in[24];
    float* out = (float*)d_out;

    float* ws = (float*)d_ws;
    size_t off = 0;
    auto alloc = [&](size_t n) { float* p = ws + off; off += n; return p; };

    float* feat_att = alloc((size_t)Bsz * Pp * Aa);   // 51 MB
    float* hWh      = alloc((size_t)Bsz * Aa);
    float* scores   = alloc((size_t)Bsz * Pp);
    float* ctx      = alloc((size_t)Bsz * Ff);
    float* ctxp     = alloc((size_t)Bsz * Hh);
    float* h0       = alloc((size_t)Bsz * Hh);        // h0,c0,h1,c1 contiguous for one memset
    float* c0       = alloc((size_t)Bsz * Hh);
    float* h1       = alloc((size_t)Bsz * Hh);
    float* c1       = alloc((size_t)Bsz * Hh);
    float* xh0      = alloc((size_t)Bsz * 1536);
    float* gates0   = alloc((size_t)Bsz * 2048);
    float* xh1      = alloc((size_t)Bsz * 1024);
    float* gates1   = alloc((size_t)Bsz * 2048);
    float* skipin   = alloc((size_t)Bsz * 1536);
    float* skipv    = alloc((size_t)Bsz * Hh);
    float* lnv      = alloc((size_t)Bsz * Hh);
    float* W0cat    = alloc((size_t)2048 * 1536);
    float* W1cat    = alloc((size_t)2048 * 1024);
    float* bias0    = alloc(2048);
    float* bias1    = alloc(2048);

    // zero initial h/c state (ws is poisoned by the harness)
    hipMemsetAsync(h0, 0, (size_t)4 * Bsz * Hh * sizeof(float), stream);

    // pack LSTM weights / biases once per launch
    pack_cat_kernel<<<(2048 * 1536 + 255) / 256, 256, 0, stream>>>(W_ih0, 1024, W_hh0, 512, W0cat, 2048);
    pack_cat_kernel<<<(2048 * 1024 + 255) / 256, 256, 0, stream>>>(W_ih1, 512,  W_hh1, 512, W1cat, 2048);
    add_vec_kernel<<<8, 256, 0, stream>>>(b_ih0, b_hh0, bias0, 2048);
    add_vec_kernel<<<8, 256, 0, stream>>>(b_ih1, b_hh1, bias1, 2048);

    // feat_att = features @ att_Wf + att_bf   [B*P, A]
    {
        dim3 g((Aa + 63) / 64, (Bsz * Pp + 63) / 64);
        gemm_wmma_f32<false, false><<<g, 128, 0, stream>>>(
            features, Ff, att_Wf, Aa, att_bf, feat_att, Aa, Bsz * Pp, Aa, Ff);
    }

    for (int t = 0; t < Tt; ++t) {
        // hWh = h1 @ att_Wh + att_bh
        {
            dim3 g((Aa + 63) / 64, (Bsz + 63) / 64);
            gemm_wmma_f32<false, false><<<g, 128, 0, stream>>>(
                h1, Hh, att_Wh, Aa, att_bh, hWh, Aa, Bsz, Aa, Hh);
        }
        att_scores_kernel<<<(Bsz * Pp) / 4, 128, 0, stream>>>(feat_att, hWh, att_v, att_bv, scores);
        softmax_ctx_kernel<<<Bsz, 256, 0, stream>>>(scores, features, ctx);
        // ctxp = relu(ctx @ proj_W + proj_b)
        {
            dim3 g((Hh + 63) / 64, (Bsz + 63) / 64);
            gemm_wmma_f32<false, true><<<g, 128, 0, stream>>>(
                ctx, Ff, proj_W, Hh, proj_b, ctxp, Hh, Bsz, Hh, Ff);
        }
        build_xh0_kernel<<<(Bsz * 1536 + 255) / 256, 256, 0, stream>>>(captions, t, emb, ctxp, h0, xh0);
        // gates0 = [x|h0] @ [W_ih0|W_hh0]^T + (b_ih0+b_hh0)
        {
            dim3 g((2048 + 63) / 64, (Bsz + 63) / 64);
            gemm_wmma_f32<true, false><<<g, 128, 0, stream>>>(
                xh0, 1536, W0cat, 1536, bias0, gates0, 2048, Bsz, 2048, 1536);
        }
        lstm_update_kernel<<<(Bsz * Hh + 255) / 256, 256, 0, stream>>>(gates0, h0, c0);
        build_xh1_kernel<<<(Bsz * 1024 + 255) / 256, 256, 0, stream>>>(h0, h1, xh1);
        // gates1 = [h0n|h1] @ [W_ih1|W_hh1]^T + (b_ih1+b_hh1)
        {
            dim3 g((2048 + 63) / 64, (Bsz + 63) / 64);
            gemm_wmma_f32<true, false><<<g, 128, 0, stream>>>(
                xh1, 1024, W1cat, 1024, bias1, gates1, 2048, Bsz, 2048, 1024);
        }
        lstm_update_kernel<<<(Bsz * Hh + 255) / 256, 256, 0, stream>>>(gates1, h1, c1);
        build_skipin_kernel<<<(Bsz * 1536 + 255) / 256, 256, 0, stream>>>(h1, captions, t, emb, ctxp, skipin);
        // skip = [h1n, x] @ skip_W + skip_b
        {
            dim3 g((Hh + 63) / 64, (Bsz + 63) / 64);
            gemm_wmma_f32<false, false><<<g, 128, 0, stream>>>(
                skipin, 1536, skip_W, Hh, skip_b, skipv, Hh, Bsz, Hh, 1536);
        }
        layernorm_kernel<<<Bsz, 256, 0, stream>>>(skipv, ln_g, ln_b, lnv);
        // out[:, t, :] = ln @ fc_W + fc_b   (ldc = T*V strides over the time axis)
        {
            dim3 g((Vv + 63) / 64, (Bsz + 63) / 64);
            gemm_wmma_f32<false, false><<<g, 128, 0, stream>>>(
                lnv, Hh, fc_W, Vv, fc_b, out + (size_t)t * Vv, Tt * Vv, Bsz, Vv, Hh);
        }
    }
}